// GCN3DEncoder_13554916786447
// MI455X (gfx1250) — compile-verified
//
#include <hip/hip_runtime.h>
#include <math.h>

// GCN3D encoder for gfx1250 (MI455X). f32 end-to-end (matches reference dtype).
// Dense GEMMs use V_WMMA_F32_16X16X4_F32: 4-wave blocks compute a 16x64 C tile,
// A panel staged in LDS (shared by all 4 waves), guard-free inner loop (all
// shapes padded so M%16==0, N%64==0, K%32==0 -> EXEC all-1s, zero divergence).
// Global accesses in the GEMM use SGPR-base + 32-bit lane index (GVS saddr +
// scale_offset form) so the hot loop has no 64-bit VGPR address arithmetic.
// Graph ops (kNN, theta/gather/max/sum, pool, global max) are fused VALU
// kernels so no (bs,v,n,s*cout) intermediate is ever materialized.

#define BS   4
#define NNB  16
#define SUP  4
#define V0   4096
#define KC   32      // K chunk staged in LDS (all layer K's are multiples of 32)

typedef __attribute__((ext_vector_type(2))) float v2f;
typedef __attribute__((ext_vector_type(8))) float v8f;

// ---------------- column-normalize a (3, C) direction matrix ----------------
__global__ void norm_dirs_kernel(const float* __restrict__ d, float* __restrict__ out, int C) {
  int c = blockIdx.x * blockDim.x + threadIdx.x;
  if (c >= C) return;
  float x = d[c], y = d[C + c], z = d[2 * C + c];
  float inv = 1.0f / fmaxf(sqrtf(x * x + y * y + z * z), 1e-12f);
  out[c] = x * inv; out[C + c] = y * inv; out[2 * C + c] = z * inv;
}

// ---------------- brute-force kNN, excludes self ----------------
// queries are rows j*qstride of the same vertex array. Register-resident sorted top-KN.
template<int KN>
__global__ void knn_kernel(const float* __restrict__ vtx, int bs, int v, int q, int qstride,
                           int* __restrict__ idx_out) {
  int t = blockIdx.x * blockDim.x + threadIdx.x;
  if (t >= bs * q) return;
  int b = t / q, j = t - b * q;
  int i = j * qstride;
  const float* V = vtx + (size_t)b * v * 3;
  float xi = V[3 * i], yi = V[3 * i + 1], zi = V[3 * i + 2];
  float bd[KN]; int bi[KN];
#pragma unroll
  for (int k = 0; k < KN; ++k) { bd[k] = 3.4e38f; bi[k] = 0; }
  for (int jj = 0; jj < v; ++jj) {
    if (jj == i) continue;
    float dx = V[3 * jj] - xi, dy = V[3 * jj + 1] - yi, dz = V[3 * jj + 2] - zi;
    float dd = dx * dx + dy * dy + dz * dz;
    if (dd < bd[KN - 1]) {
      bd[KN - 1] = dd; bi[KN - 1] = jj;
#pragma unroll
      for (int k = KN - 1; k > 0; --k) {
        if (bd[k] < bd[k - 1]) {   // strict < keeps stable (lower-index-first) order on ties
          float td = bd[k]; bd[k] = bd[k - 1]; bd[k - 1] = td;
          int   ti = bi[k]; bi[k] = bi[k - 1]; bi[k - 1] = ti;
        }
      }
    }
  }
  int* o = idx_out + (size_t)t * KN;
#pragma unroll
  for (int k = 0; k < KN; ++k) o[k] = bi[k];
}

// ---------------- conv_surface: fm0[b,i,c] = relu(sum_s max_n relu(ndn . sdn[:,s*ksz+c])) ----
__global__ void conv_surface_kernel(const int* __restrict__ idx, const float* __restrict__ vtx,
                                    const float* __restrict__ sdn, float* __restrict__ out,
                                    int bs, int v, int ksz) {
  int t = blockIdx.x * blockDim.x + threadIdx.x;
  int total = bs * v * ksz;
  if (t >= total) return;
  int c = t % ksz;
  int i = (t / ksz) % v;
  int b = t / (ksz * v);
  int C = SUP * ksz;
  const float* V = vtx + (size_t)b * v * 3;
  const int* nb = idx + ((size_t)b * v + i) * NNB;
  float xi = V[3 * i], yi = V[3 * i + 1], zi = V[3 * i + 2];
  float m[SUP];
#pragma unroll
  for (int s = 0; s < SUP; ++s) m[s] = 0.0f;   // theta >= 0, so max over n >= 0
  for (int n = 0; n < NNB; ++n) {
    int jn = nb[n];
    float dx = V[3 * jn] - xi, dy = V[3 * jn + 1] - yi, dz = V[3 * jn + 2] - zi;
    float inv = 1.0f / fmaxf(sqrtf(dx * dx + dy * dy + dz * dz), 1e-12f);
    dx *= inv; dy *= inv; dz *= inv;
#pragma unroll
    for (int s = 0; s < SUP; ++s) {
      int col = s * ksz + c;
      float th = dx * sdn[col] + dy * sdn[C + col] + dz * sdn[2 * C + col];
      th = fmaxf(th, 0.0f);
      m[s] = fmaxf(m[s], th);
    }
  }
  float r = 0.0f;
#pragma unroll
  for (int s = 0; s < SUP; ++s) r += m[s];
  out[((size_t)b * v + i) * ksz + c] = fmaxf(r, 0.0f);
}

// ---------------- conv_layer aggregation ----------------
// out[b,i,c] = (relu?)( feat[b,i,c] + sum_s max_n relu(ndn.sdn[:,s*oc+c]) * feat[b,idx,oc+s*oc+c] )
__global__ void conv_agg_kernel(const int* __restrict__ idx, const float* __restrict__ vtx,
                                const float* __restrict__ feat, const float* __restrict__ sdn,
                                float* __restrict__ out, int bs, int v, int out_ch, int do_relu) {
  int t = blockIdx.x * blockDim.x + threadIdx.x;
  int total = bs * v * out_ch;
  if (t >= total) return;
  int c = t % out_ch;
  int i = (t / out_ch) % v;
  int b = t / (out_ch * v);
  int fld = (SUP + 1) * out_ch;
  int C = SUP * out_ch;
  const float* V = vtx + (size_t)b * v * 3;
  const float* F = feat + (size_t)b * v * fld;
  const int* nb = idx + ((size_t)b * v + i) * NNB;
  float xi = V[3 * i], yi = V[3 * i + 1], zi = V[3 * i + 2];
  float m[SUP];
#pragma unroll
  for (int s = 0; s < SUP; ++s) m[s] = -3.4e38f;   // theta*support can be negative
  for (int n = 0; n < NNB; ++n) {
    int jn = nb[n];
    float dx = V[3 * jn] - xi, dy = V[3 * jn + 1] - yi, dz = V[3 * jn + 2] - zi;
    float inv = 1.0f / fmaxf(sqrtf(dx * dx + dy * dy + dz * dz), 1e-12f);
    dx *= inv; dy *= inv; dz *= inv;
    const float* sp = F + (size_t)jn * fld + out_ch;
#pragma unroll
    for (int s = 0; s < SUP; ++s) {
      int col = s * out_ch + c;
      float th = dx * sdn[col] + dy * sdn[C + col] + dz * sdn[2 * C + col];
      th = fmaxf(th, 0.0f);
      m[s] = fmaxf(m[s], th * sp[col]);
    }
  }
  float r = F[(size_t)i * fld + c];
#pragma unroll
  for (int s = 0; s < SUP; ++s) r += m[s];
  if (do_relu) r = fmaxf(r, 0.0f);
  out[((size_t)b * v + i) * out_ch + c] = r;
}

// ---------------- pool: fm_out[b,j,c] = max over 4 kNN of fm_in; vtx_out[b,j] = vtx_in[b,j*pr] --
__global__ void pool_kernel(const int* __restrict__ idx4, const float* __restrict__ vtx_in,
                            const float* __restrict__ fm_in, float* __restrict__ vtx_out,
                            float* __restrict__ fm_out, int bs, int v_in, int q, int pr, int ch) {
  int t = blockIdx.x * blockDim.x + threadIdx.x;
  int total = bs * q * ch;
  if (t >= total) return;
  int c = t % ch;
  int j = (t / ch) % q;
  int b = t / (ch * q);
  const int* nb = idx4 + ((size_t)b * q + j) * 4;
  float m = -3.4e38f;
#pragma unroll
  for (int n = 0; n < 4; ++n)
    m = fmaxf(m, fm_in[((size_t)b * v_in + nb[n]) * ch + c]);
  fm_out[((size_t)b * q + j) * ch + c] = m;
  if (c < 3)
    vtx_out[((size_t)b * q + j) * 3 + c] = vtx_in[((size_t)b * v_in + (size_t)j * pr) * 3 + c];
}

// ---------------- global max over vertices, zero-padded to rows_out rows ----------------
__global__ void vmax_pad_kernel(const float* __restrict__ fm, float* __restrict__ out,
                                int bs, int v, int ch, int rows_out) {
  int t = blockIdx.x * blockDim.x + threadIdx.x;
  if (t >= rows_out * ch) return;
  int c = t % ch, b = t / ch;
  float m = 0.0f;
  if (b < bs) {
    m = -3.4e38f;
    for (int i = 0; i < v; ++i)
      m = fmaxf(m, fm[((size_t)b * v + i) * ch + c]);
  }
  out[(size_t)b * ch + c] = m;
}

// ---------------- f32 WMMA GEMM: C = A(MxK) * B(KxN) + bias, row-major ----------------
// Requirements (arranged by the launcher): M%16==0, N%64==0, K%32==0.
// Block = 128 threads (4 wave32) -> 16x64 C tile; wave w owns N columns
// [blockIdx.x*64 + 16w, +16). A panel chunk (16xKC) staged in LDS, stride 33
// (gcd(33,64)=1 -> conflict-free across the 64 LDS banks). Inner loop:
// 8 x V_WMMA_F32_16X16X4_F32 per chunk; B and C addressed as SGPR base +
// 32-bit element index (GVS saddr + scale_offset), single v_add per step.
// Frag layout (16x16x4 f32): A lanes0-15 row=l K={k,k+1}, lanes16-31 K={k+2,k+3};
// B lanes = N col, vgpr pair = K (lo/hi half offset by 2); D vgpr r -> row
// m0+r+8*half, col n0+l.
__global__ void __launch_bounds__(128)
gemm_wmma_f32(const float* __restrict__ A, const float* __restrict__ B,
              const float* __restrict__ bias, float* __restrict__ Cm,
              int M, int K, int N) {
  __shared__ float As[16 * 33];
  const int tid  = threadIdx.x;
  const int lane = tid & 31;
  const int wave = tid >> 5;
  const int m0 = blockIdx.y * 16;
  const int n0 = blockIdx.x * 64 + wave * 16;
  const int half = lane >> 4;
  const int l = lane & 15;

  const unsigned uN = (unsigned)N, uK = (unsigned)K;
  const unsigned abase = (unsigned)m0 * uK;                         // A panel base (elements)
  const unsigned bidx0 = (unsigned)(2 * half) * uN + (unsigned)(n0 + l);
  const unsigned stepB = 4u * uN;
  v8f acc = {};

  for (int kc = 0; kc < K; kc += KC) {
    // cooperative coalesced load of A[m0..m0+15][kc..kc+KC) into LDS
#pragma unroll
    for (int i2 = 0; i2 < (16 * KC) / 128; ++i2) {
      int e = tid + 128 * i2;
      int r = e >> 5, cc = e & 31;
      As[r * 33 + cc] = A[abase + (unsigned)r * uK + (unsigned)(kc + cc)];
    }
    __syncthreads();
    unsigned bidx = bidx0 + (unsigned)kc * uN;
    const float* Ar = &As[l * 33 + 2 * half];
#pragma unroll
    for (int kk = 0; kk < KC; kk += 4) {
      v2f a, bf;
      a.x = Ar[kk];
      a.y = Ar[kk + 1];
      bf.x = B[bidx];
      bf.y = B[bidx + uN];
      bidx += stepB;
      acc = __builtin_amdgcn_wmma_f32_16x16x4_f32(
          /*neg_a=*/false, a, /*neg_b=*/false, bf,
          /*c_mod=*/(short)0, acc, /*reuse_a=*/false, /*reuse_b=*/false);
    }
    __syncthreads();   // protect LDS before next chunk overwrites it
  }

  const int ocol = n0 + l;
  const float bv = bias[ocol];
  unsigned cidx = (unsigned)(m0 + 8 * half) * uN + (unsigned)ocol;
#pragma unroll
  for (int r = 0; r < 8; ++r) {
    Cm[cidx] = acc[r] + bv;
    cidx += uN;
  }
}

extern "C" void kernel_launch(void* const* d_in, const int* in_sizes, int n_in,
                              void* d_out, int out_size, void* d_ws, size_t ws_size,
                              hipStream_t stream) {
  const float* vertices = (const float*)d_in[0];
  const float* d0 = (const float*)d_in[1];
  const float* w1 = (const float*)d_in[2];
  const float* b1 = (const float*)d_in[3];
  const float* d1 = (const float*)d_in[4];
  const float* w2 = (const float*)d_in[5];
  const float* b2 = (const float*)d_in[6];
  const float* d2 = (const float*)d_in[7];
  const float* w3 = (const float*)d_in[8];
  const float* b3 = (const float*)d_in[9];
  const float* d3 = (const float*)d_in[10];
  const float* w4 = (const float*)d_in[11];
  const float* b4 = (const float*)d_in[12];
  const float* d4 = (const float*)d_in[13];
  const float* lw = (const float*)d_in[14];
  const float* lb = (const float*)d_in[15];

  char* ws = (char*)d_ws;
  size_t off = 0;
  auto alloc = [&](size_t bytes) -> void* {
    void* p = ws + off;
    off += (bytes + 255) & ~(size_t)255;
    return p;
  };
  float* sdn0 = (float*)alloc((size_t)3 * 128  * 4);
  float* sdn1 = (float*)alloc((size_t)3 * 256  * 4);
  float* sdn2 = (float*)alloc((size_t)3 * 512  * 4);
  float* sdn3 = (float*)alloc((size_t)3 * 1024 * 4);
  float* sdn4 = (float*)alloc((size_t)3 * 4096 * 4);
  int*   idx16 = (int*)alloc((size_t)BS * V0 * NNB * 4);   // 1 MB, reused per stage
  int*   idx4  = (int*)alloc((size_t)BS * 1024 * 4 * 4);
  float* vtx2  = (float*)alloc((size_t)BS * 1024 * 3 * 4);
  float* vtx3  = (float*)alloc((size_t)BS * 256  * 3 * 4);
  float* fmA   = (float*)alloc((size_t)BS * V0 * 64  * 4); // 4 MB ping
  float* fmB   = (float*)alloc((size_t)BS * V0 * 64  * 4); // 4 MB pong
  float* featb = (float*)alloc((size_t)BS * V0 * 320 * 4); // 20 MB (== BS*256*5120 for layer 4)
  float* fmax  = (float*)alloc((size_t)16 * 1024 * 4);     // padded to 16 rows
  float* pout  = (float*)alloc((size_t)16 * 512  * 4);     // padded final GEMM output
  (void)ws_size; (void)in_sizes; (void)n_in; (void)out_size;

  const int TB = 256;
  auto cdiv = [](int a, int b) { return (a + b - 1) / b; };
  auto gemm = [&](const float* A, const float* Bw, const float* bias, float* Cm,
                  int M, int K, int N) {    // M%16==0, N%64==0, K%32==0 guaranteed
    dim3 grid(N / 64, M / 16);
    gemm_wmma_f32<<<grid, 128, 0, stream>>>(A, Bw, bias, Cm, M, K, N);
  };

  // normalized support directions
  norm_dirs_kernel<<<cdiv(128,  TB), TB, 0, stream>>>(d0, sdn0, 128);
  norm_dirs_kernel<<<cdiv(256,  TB), TB, 0, stream>>>(d1, sdn1, 256);
  norm_dirs_kernel<<<cdiv(512,  TB), TB, 0, stream>>>(d2, sdn2, 512);
  norm_dirs_kernel<<<cdiv(1024, TB), TB, 0, stream>>>(d3, sdn3, 1024);
  norm_dirs_kernel<<<cdiv(4096, TB), TB, 0, stream>>>(d4, sdn4, 4096);

  // ---- stage 1: 4096 vertices ----
  knn_kernel<16><<<cdiv(BS * V0, 128), 128, 0, stream>>>(vertices, BS, V0, V0, 1, idx16);
  conv_surface_kernel<<<cdiv(BS * V0 * 32, TB), TB, 0, stream>>>(idx16, vertices, sdn0, fmA, BS, V0, 32);
  gemm(fmA, w1, b1, featb, BS * V0, 32, 320);
  conv_agg_kernel<<<cdiv(BS * V0 * 64, TB), TB, 0, stream>>>(idx16, vertices, featb, sdn1, fmB, BS, V0, 64, 1);

  // ---- pool 1: 4096 -> 1024 (deterministic stride-4 row selection in lieu of jax permutation)
  knn_kernel<4><<<cdiv(BS * 1024, 128), 128, 0, stream>>>(vertices, BS, V0, 1024, 4, idx4);
  pool_kernel<<<cdiv(BS * 1024 * 64, TB), TB, 0, stream>>>(idx4, vertices, fmB, vtx2, fmA, BS, V0, 1024, 4, 64);

  // ---- stages 2 & 3: 1024 vertices ----
  knn_kernel<16><<<cdiv(BS * 1024, 128), 128, 0, stream>>>(vtx2, BS, 1024, 1024, 1, idx16);
  gemm(fmA, w2, b2, featb, BS * 1024, 64, 640);
  conv_agg_kernel<<<cdiv(BS * 1024 * 128, TB), TB, 0, stream>>>(idx16, vtx2, featb, sdn2, fmB, BS, 1024, 128, 1);
  gemm(fmB, w3, b3, featb, BS * 1024, 128, 1280);
  conv_agg_kernel<<<cdiv(BS * 1024 * 256, TB), TB, 0, stream>>>(idx16, vtx2, featb, sdn3, fmA, BS, 1024, 256, 1);

  // ---- pool 2: 1024 -> 256 ----
  knn_kernel<4><<<cdiv(BS * 256, 128), 128, 0, stream>>>(vtx2, BS, 1024, 256, 4, idx4);
  pool_kernel<<<cdiv(BS * 256 * 256, TB), TB, 0, stream>>>(idx4, vtx2, fmA, vtx3, fmB, BS, 1024, 256, 4, 256);

  // ---- stage 4: 256 vertices (no relu) ----
  knn_kernel<16><<<cdiv(BS * 256, 128), 128, 0, stream>>>(vtx3, BS, 256, 256, 1, idx16);
  gemm(fmB, w4, b4, featb, BS * 256, 256, 5120);
  conv_agg_kernel<<<cdiv(BS * 256 * 1024, TB), TB, 0, stream>>>(idx16, vtx3, featb, sdn4, fmA, BS, 256, 1024, 0);

  // ---- global max (zero-padded to 16 rows) + final linear (M=16) ----
  vmax_pad_kernel<<<cdiv(16 * 1024, TB), TB, 0, stream>>>(fmA, fmax, BS, 256, 1024, 16);
  gemm(fmax, lw, lb, pout, 16, 1024, 512);
  // rows 0..3 of the padded 16x512 result are the real output (contiguous)
  hipMemcpyAsync(d_out, pout, (size_t)BS * 512 * 4, hipMemcpyDeviceToDevice, stream);
}